// Model_40973988004218
// MI455X (gfx1250) — compile-verified
//
#include <hip/hip_runtime.h>
#include <hip/hip_bf16.h>
#include <math.h>

// ---------------------------------------------------------------------------
// Hyperbolic seq2seq forecaster for MI455X (gfx1250, wave32, WMMA).
// Latency-bound serial scan -> single resident kernel per 16-row batch slice,
// f32 WMMA (V_WMMA_F32_16X16X4_F32) for every matmul, weights held in LDS.
// Key scan identity: tv_{s+1} = logmap0(zn_s) = the value already needed for
// x_pred_s, so logmap0 is computed once per step and the z state is never
// materialized.
// ---------------------------------------------------------------------------

#define B_   128
#define T_   2048
#define C_   128
#define E_   32
#define H_   64
#define PRED 336
#define EPSF 1e-6f

typedef __attribute__((ext_vector_type(2))) float v2f;
typedef __attribute__((ext_vector_type(8))) float v8f;

__device__ __forceinline__ v8f wmma4(v2f a, v2f b, v8f c) {
  // D = A(16x4) * B(4x16) + C(16x16), all f32.
  return __builtin_amdgcn_wmma_f32_16x16x4_f32(
      /*neg_a=*/false, a, /*neg_b=*/false, b,
      /*c_mod=*/(short)0, c, /*reuse_a=*/false, /*reuse_b=*/false);
}

// A-matrix 16x4 fragment (row-major source, stride lda).
// Lanes 0-15: (M=lane, K=kk+0 / kk+1); lanes 16-31: (M=lane-16, K=kk+2 / kk+3).
__device__ __forceinline__ v2f load_a(const float* A, int lda, int row0, int kk, int lane) {
  const int half = lane >> 4, lr = lane & 15;
  const float* p = A + (size_t)(row0 + lr) * lda + kk + 2 * half;
  v2f r; r.x = p[0]; r.y = p[1];
  return r;
}

// B-matrix 4x16 fragment (row-major source K x N, stride ldb).
// VGPR0: rows kk+0 (lanes 0-15) / kk+2 (lanes 16-31); VGPR1: kk+1 / kk+3.
__device__ __forceinline__ v2f load_b(const float* Bm, int ldb, int kk, int col0, int lane) {
  const int half = lane >> 4, lc = lane & 15;
  const float* p = Bm + (size_t)(kk + 2 * half) * ldb + col0 + lc;
  v2f r; r.x = p[0]; r.y = p[ldb];
  return r;
}

__device__ __forceinline__ float sigmoidf_(float x) { return 1.0f / (1.0f + expf(-x)); }

// ---------------------------------------------------------------------------
// Kernel 1: moving-average trend (k=7, edge pad) on channel 0 only.
// ---------------------------------------------------------------------------
__global__ __launch_bounds__(256) void trend_kernel(const float* __restrict__ x,
                                                    float* __restrict__ t0,
                                                    float* __restrict__ r0) {
  const int t = blockIdx.x * 256 + threadIdx.x;
  const int b = blockIdx.y;
  if (t >= T_) return;
  const float* xb = x + (size_t)b * T_ * C_;
  __builtin_prefetch(&xb[(size_t)t * C_], 0, 0);  // global_prefetch_b8
  float s = 0.0f;
#pragma unroll
  for (int j = -3; j <= 3; ++j) {
    int tc = t + j;
    tc = tc < 0 ? 0 : (tc > T_ - 1 ? T_ - 1 : tc);
    s += xb[(size_t)tc * C_];
  }
  const float tr = s * (1.0f / 7.0f);
  t0[b * T_ + t] = tr;
  r0[b * T_ + t] = xb[(size_t)t * C_] - tr;
}

// ---------------------------------------------------------------------------
// Kernel 2: v = sum of tanh(GEMMs), z0' = logmap0(expmap0(v)) row factors.
// We store tv0 = logmap0(expmap0(v)) directly (what the scan actually needs).
// 8 blocks (16 rows) x 8 waves: wave = (matrix m, col-tile ct).
// K=2048 via f32 WMMA with 4 independent accumulators (chain depth 128).
// ---------------------------------------------------------------------------
__global__ __launch_bounds__(256) void encoder_kernel(
    const float* __restrict__ t0, const float* __restrict__ r0,
    const float* __restrict__ We_trend, const float* __restrict__ We_weekly,
    const float* __restrict__ We_daily, const float* __restrict__ We_resid,
    const float* __restrict__ be_trend, const float* __restrict__ be_weekly,
    const float* __restrict__ be_daily, const float* __restrict__ be_resid,
    float* __restrict__ tv0out) {
  __shared__ float vsum[16 * E_];
  const int tid = threadIdx.x;
  for (int i = tid; i < 16 * E_; i += 256) vsum[i] = 0.0f;
  __syncthreads();

  const int wv = tid >> 5, lane = tid & 31;
  const int m = wv >> 1, ct = wv & 1;
  const float* A = (m == 0) ? t0 : r0;
  const float* W = (m == 0) ? We_trend : (m == 1) ? We_weekly : (m == 2) ? We_daily : We_resid;
  const float* bias = (m == 0) ? be_trend : (m == 1) ? be_weekly : (m == 2) ? be_daily : be_resid;
  const int row0 = blockIdx.x * 16;
  const int col0 = ct * 16;

  v8f c0 = {}, c1 = {}, c2 = {}, c3 = {};
  for (int kk = 0; kk < T_; kk += 16) {
    c0 = wmma4(load_a(A, T_, row0, kk + 0,  lane), load_b(W, E_, kk + 0,  col0, lane), c0);
    c1 = wmma4(load_a(A, T_, row0, kk + 4,  lane), load_b(W, E_, kk + 4,  col0, lane), c1);
    c2 = wmma4(load_a(A, T_, row0, kk + 8,  lane), load_b(W, E_, kk + 8,  col0, lane), c2);
    c3 = wmma4(load_a(A, T_, row0, kk + 12, lane), load_b(W, E_, kk + 12, col0, lane), c3);
  }

  const int half = lane >> 4, lc = lane & 15;
  const int col = col0 + lc;
#pragma unroll
  for (int i = 0; i < 8; ++i) {
    const int row = i + 8 * half;
    const float acc = c0[i] + c1[i] + c2[i] + c3[i];
    atomicAdd(&vsum[row * E_ + col], tanhf(acc + bias[col]));
  }
  __syncthreads();

  // tv0 = logmap0(expmap0(v)) per row; ||scale*v|| = scale*||v||.
  if (tid < 16) {
    const int row = tid, grow = row0 + row;
    float n2 = 0.0f;
    for (int j = 0; j < E_; ++j) { float v = vsum[row * E_ + j]; n2 += v * v; }
    const float r = sqrtf(n2);
    const float x0 = coshf(r);
    const float scale = (r > EPSF) ? (sinhf(r) / fmaxf(r, EPSF)) : 1.0f;
    const float rr = acoshf(fmaxf(x0, 1.0f));
    const float fac = rr * scale / fmaxf(scale * r, EPSF);
    for (int j = 0; j < E_; ++j) tv0out[grow * E_ + j] = fac * vsum[row * E_ + j];
  }
}

// ---------------------------------------------------------------------------
// Kernel 3: 336-step hyperbolic scan, fully resident.
// 8 blocks x 128 threads (4 waves); each block owns 16 independent batch rows.
// Per step (5 barriers):
//   phase A: out_{s-1} = stv@Wr + br  AND  u_s = gates(stv)   [all reads of stv]
//   phase B: vn = stv + u@Wd_out + b
//   phase C: stv <- logmap0(expmap0(vn))   [parallel row reduction]
// ---------------------------------------------------------------------------
__global__ __launch_bounds__(128) void scan_kernel(
    const float* __restrict__ tv0,
    const float* __restrict__ Wd_in, const float* __restrict__ bd_in,
    const float* __restrict__ Wd_gate, const float* __restrict__ bd_gate,
    const float* __restrict__ Wd_out, const float* __restrict__ bd_out,
    const float* __restrict__ Wr, const float* __restrict__ br,
    float* __restrict__ out) {
  __shared__ float sWin[E_ * H_], sWg[E_ * H_], sWo[H_ * E_], sWrm[E_ * C_];
  __shared__ float sbin[H_], sbg[H_], sbo[E_], sbr[C_];
  __shared__ float stv[16 * E_];        // tangent state (doubles as logmap0(zn))
  __shared__ float su[16 * H_];         // gated hidden
  __shared__ float svn[16 * E_];        // tangent update
  __shared__ float part[16 * 8];        // per-row partial squared sums
  __shared__ float fac[16];             // per-row log∘exp scale factor

  const int tid = threadIdx.x;
  const int row0 = blockIdx.x * 16;

  for (int i = tid; i < E_ * H_; i += 128) { sWin[i] = Wd_in[i]; sWg[i] = Wd_gate[i]; }
  for (int i = tid; i < H_ * E_; i += 128) sWo[i] = Wd_out[i];
  for (int i = tid; i < E_ * C_; i += 128) sWrm[i] = Wr[i];
  if (tid < H_) { sbin[tid] = bd_in[tid]; sbg[tid] = bd_gate[tid]; }
  if (tid < E_) sbo[tid] = bd_out[tid];
  if (tid < C_) sbr[tid] = br[tid];
  for (int i = tid; i < 16 * E_; i += 128) stv[i] = tv0[row0 * E_ + i];
  __syncthreads();

  const int wv = tid >> 5, lane = tid & 31;
  const int half = lane >> 4, lc = lane & 15;
  const int prow = tid >> 3, pk = tid & 7;  // reduction mapping: 8 threads/row

  for (int s = 0; s < PRED; ++s) {
    // ---- phase A: x_pred_{s-1} (if any) and u_s, both from stv ----
    if (s > 0) {
#pragma unroll
      for (int tpart = 0; tpart < 2; ++tpart) {
        const int col0 = (wv + 4 * tpart) * 16;  // 4 waves x 2 tiles -> C_=128
        v8f c = {};
#pragma unroll
        for (int kk = 0; kk < E_; kk += 4)
          c = wmma4(load_a(stv, E_, 0, kk, lane), load_b(sWrm, C_, kk, col0, lane), c);
        const int col = col0 + lc;
#pragma unroll
        for (int i = 0; i < 8; ++i) {
          const int grow = row0 + i + 8 * half;
          out[((size_t)grow * PRED + (s - 1)) * C_ + col] = c[i] + sbr[col];
        }
      }
    }
    {
      const int col0 = wv * 16;  // 4 waves cover H_=64
      v8f ci = {}, cg = {};
#pragma unroll
      for (int kk = 0; kk < E_; kk += 4) {
        const v2f a = load_a(stv, E_, 0, kk, lane);
        ci = wmma4(a, load_b(sWin, H_, kk, col0, lane), ci);
        cg = wmma4(a, load_b(sWg, H_, kk, col0, lane), cg);
      }
      const int col = col0 + lc;
#pragma unroll
      for (int i = 0; i < 8; ++i) {
        const int row = i + 8 * half;
        const float xi = ci[i] + sbin[col];
        const float xg = cg[i] + sbg[col];
        su[row * H_ + col] = (xi * sigmoidf_(xi)) * sigmoidf_(xg);
      }
    }
    __syncthreads();

    // ---- phase B: vn = tv + u@Wd_out + b  (waves 0,1 cover E_=32) ----
    if (wv < 2) {
      const int col0 = wv * 16;
      v8f c = {};
#pragma unroll
      for (int kk = 0; kk < H_; kk += 4)
        c = wmma4(load_a(su, H_, 0, kk, lane), load_b(sWo, E_, kk, col0, lane), c);
      const int col = col0 + lc;
#pragma unroll
      for (int i = 0; i < 8; ++i) {
        const int row = i + 8 * half;
        svn[row * E_ + col] = c[i] + stv[row * E_ + col] + sbo[col];
      }
    }
    __syncthreads();

    // ---- phase C: stv <- logmap0(expmap0(vn)) ----
    {
      float p2 = 0.0f;
#pragma unroll
      for (int j = 0; j < 4; ++j) {
        const float v = svn[prow * E_ + pk * 4 + j];
        p2 += v * v;
      }
      part[prow * 8 + pk] = p2;
    }
    __syncthreads();
    if (tid < 16) {
      const int row = tid;
      float n2 = 0.0f;
#pragma unroll
      for (int k = 0; k < 8; ++k) n2 += part[row * 8 + k];
      const float r = sqrtf(n2);
      const float x0 = coshf(r);
      const float scale = (r > EPSF) ? (sinhf(r) / fmaxf(r, EPSF)) : 1.0f;
      const float rr = acoshf(fmaxf(x0, 1.0f));
      fac[row] = rr * scale / fmaxf(scale * r, EPSF);
    }
    __syncthreads();
    {
      const float f = fac[prow];
#pragma unroll
      for (int j = 0; j < 4; ++j) {
        const int idx = prow * E_ + pk * 4 + j;
        stv[idx] = f * svn[idx];
      }
    }
    __syncthreads();
  }

  // ---- final output row: x_pred_{335} from the last stv ----
#pragma unroll
  for (int tpart = 0; tpart < 2; ++tpart) {
    const int col0 = (wv + 4 * tpart) * 16;
    v8f c = {};
#pragma unroll
    for (int kk = 0; kk < E_; kk += 4)
      c = wmma4(load_a(stv, E_, 0, kk, lane), load_b(sWrm, C_, kk, col0, lane), c);
    const int col = col0 + lc;
#pragma unroll
    for (int i = 0; i < 8; ++i) {
      const int grow = row0 + i + 8 * half;
      out[((size_t)grow * PRED + (PRED - 1)) * C_ + col] = c[i] + sbr[col];
    }
  }
}

// ---------------------------------------------------------------------------
extern "C" void kernel_launch(void* const* d_in, const int* in_sizes, int n_in,
                              void* d_out, int out_size, void* d_ws, size_t ws_size,
                              hipStream_t stream) {
  const float* x         = (const float*)d_in[0];
  const float* We_trend  = (const float*)d_in[1];
  const float* be_trend  = (const float*)d_in[2];
  const float* We_weekly = (const float*)d_in[3];
  const float* be_weekly = (const float*)d_in[4];
  const float* We_daily  = (const float*)d_in[5];
  const float* be_daily  = (const float*)d_in[6];
  const float* We_resid  = (const float*)d_in[7];
  const float* be_resid  = (const float*)d_in[8];
  const float* Wd_in     = (const float*)d_in[9];
  const float* bd_in     = (const float*)d_in[10];
  const float* Wd_gate   = (const float*)d_in[11];
  const float* bd_gate   = (const float*)d_in[12];
  const float* Wd_out    = (const float*)d_in[13];
  const float* bd_out    = (const float*)d_in[14];
  const float* Wr        = (const float*)d_in[15];
  const float* br        = (const float*)d_in[16];

  float* ws  = (float*)d_ws;
  float* t0  = ws;                       // B*T
  float* r0  = ws + (size_t)B_ * T_;     // B*T
  float* tv0 = ws + (size_t)2 * B_ * T_; // B*E

  dim3 g1(T_ / 256, B_);
  trend_kernel<<<g1, 256, 0, stream>>>(x, t0, r0);
  encoder_kernel<<<B_ / 16, 256, 0, stream>>>(t0, r0, We_trend, We_weekly, We_daily,
                                              We_resid, be_trend, be_weekly, be_daily,
                                              be_resid, tv0);
  scan_kernel<<<B_ / 16, 128, 0, stream>>>(tv0, Wd_in, bd_in, Wd_gate, bd_gate,
                                           Wd_out, bd_out, Wr, br, (float*)d_out);
}